// TransEncoder_2353642078841
// MI455X (gfx1250) — compile-verified
//
#include <hip/hip_runtime.h>
#include <math.h>

// ---------------------------------------------------------------------------
// TransformerConv x2 on MI455X (gfx1250, wave32).
//  - Projections: V_WMMA_F32_16X16X4_F32 (fp32 matrix pipe, matches fp32 ref),
//    operands DMA'd into LDS by the Tensor Data Mover with per-row padding
//    (pad_enable) so the WMMA feed loop is LDS-bank-conflict-free.
//  - Edge softmax/aggregation: L2-resident gather/scatter with fp32 atomics.
// ---------------------------------------------------------------------------

#define DIN  128
#define DOUT 64
#define LDS_PAD 4   // dwords of skew per row: (K+4) % 64 != 0 -> conflict-free

typedef __attribute__((ext_vector_type(2))) float        v2f;
typedef __attribute__((ext_vector_type(8))) float        v8f;
typedef __attribute__((ext_vector_type(4))) unsigned int v4u;
typedef __attribute__((ext_vector_type(8))) int          v8i;
typedef __attribute__((ext_vector_type(4))) int          v4i;

// ---------------------------------------------------------------------------
// TDM: DMA a 2D tile (rows x rowlen fp32, global row stride = rowlen elems)
// from global memory into LDS at byte offset lds_addr, inserting LDS_PAD
// dwords of LDS padding after every rowlen dwords (rowlen must be a power of
// two: pad_interval code = log2(rowlen)-1). D# layout per ISA 8.3/8.4.
// Wave-level op (EXEC ignored); caller must s_wait_tensorcnt + barrier.
// ---------------------------------------------------------------------------
#if __has_builtin(__builtin_amdgcn_tensor_load_to_lds)
#define HAVE_TDM 1
static __device__ __forceinline__
void tdm_load_2d_padded(unsigned int lds_addr, const void* gptr, int rowlen, int rows)
{
  const unsigned long long ga = (unsigned long long)gptr;
  const unsigned int icode = (unsigned int)(31 - __builtin_clz((unsigned int)rowlen)) - 1u;
  v4u g0;
  g0[0] = 1u;                                             // count=1, user mode
  g0[1] = lds_addr;                                       // lds_addr [63:32]
  g0[2] = (unsigned int)(ga & 0xFFFFFFFFu);               // global_addr[31:0]
  g0[3] = (unsigned int)((ga >> 32) & 0x1FFFFFFu)         // global_addr[56:32]
        | (2u << 30);                                     // type=2 ("image")
  v8i g1;
  g1[0] = (int)((2u << 16)                                // data_size=4B
              | (1u << 20)                                // pad_enable
              | (icode << 22)                             // pad_interval: rowlen dwords
              | ((unsigned int)(LDS_PAD - 1) << 25));     // pad_amount: LDS_PAD dwords
  g1[1] = (int)((unsigned int)rowlen << 16);              // tensor_dim0[15:0]
  g1[2] = (int)((unsigned int)rows   << 16);              // tensor_dim1[15:0]
  g1[3] = (int)((unsigned int)rowlen << 16);              // tile_dim0
  g1[4] = (int)((unsigned int)rows & 0xFFFFu);            // tile_dim1 (tile_dim2=0)
  g1[5] = rowlen;                                         // tensor_dim0_stride lo
  g1[6] = 0;                                              // stride hi / dim1_stride
  g1[7] = 0;
  const v4i z = {0, 0, 0, 0};                             // groups 2/3: dims unused
#if defined(__clang_major__) && (__clang_major__ >= 23)
  const v8i z8 = {0, 0, 0, 0, 0, 0, 0, 0};
  __builtin_amdgcn_tensor_load_to_lds(g0, g1, z, z, z8, 0);
#else
  __builtin_amdgcn_tensor_load_to_lds(g0, g1, z, z, 0);
#endif
}
#else
#define HAVE_TDM 0
#endif

static __device__ __forceinline__ void wait_tensorcnt0()
{
#if __has_builtin(__builtin_amdgcn_s_wait_tensorcnt)
  __builtin_amdgcn_s_wait_tensorcnt(0);
#else
  asm volatile("s_wait_tensorcnt 0x0" ::: "memory");
#endif
}

// ---------------------------------------------------------------------------
// init per-node softmax state: m = -inf, s = 0
// ---------------------------------------------------------------------------
__global__ void init_ms_kernel(float* __restrict__ m, float* __restrict__ s, int n) {
  int i = blockIdx.x * blockDim.x + threadIdx.x;
  if (i < n) { m[i] = -INFINITY; s[i] = 0.0f; }
}

// ---------------------------------------------------------------------------
// Fused 4-way projection: O = X @ W^T + b for (Wq,Wk,Wv,Wskip) selected by
// blockIdx.y. 128-thread block = 4 waves; wave w computes the 16x16 tile at
// cols [16w,16w+16) of a 16-row stripe, K stepped by 4 with
// v_wmma_f32_16x16x4_f32. X tile (16xK) and W (64xK) are DMA'd into LDS with
// a 4-dword row skew (bank-conflict-free feed); WMMA loop reads LDS only.
// ---------------------------------------------------------------------------
__global__ __launch_bounds__(128)
void proj4_wmma_kernel(const float* __restrict__ X, int K,
                       const float* __restrict__ Wq, const float* __restrict__ bq, float* __restrict__ Oq,
                       const float* __restrict__ Wk, const float* __restrict__ bk, float* __restrict__ Ok,
                       const float* __restrict__ Wv, const float* __restrict__ bv, float* __restrict__ Ov,
                       const float* __restrict__ Ws, const float* __restrict__ bs, float* __restrict__ Os)
{
  __shared__ float smem[80 * (128 + LDS_PAD)];  // xs[16 rows] | wls[64 rows], ~41.3KB
  const int KP = K + LDS_PAD;                   // skewed LDS row stride (dwords)
  float* xs  = smem;
  float* wls = smem + 16 * KP;

  const int lane = threadIdx.x & 31;
  const int wave = threadIdx.x >> 5;
  const int half = lane >> 4;             // 0: lanes 0-15, 1: lanes 16-31
  const int l15  = lane & 15;
  const int row_base = blockIdx.x * 16;
  const int col = wave * 16 + l15;        // output column (0..63)

  const float* W; const float* B; float* O;
  switch (blockIdx.y) {
    case 0:  W = Wq; B = bq; O = Oq; break;
    case 1:  W = Wk; B = bk; O = Ok; break;
    case 2:  W = Wv; B = bv; O = Ov; break;
    default: W = Ws; B = bs; O = Os; break;
  }

#if HAVE_TDM
  if (wave == 0) {
    // flat shared pointer truncated to 32 bits == LDS byte address (ISA 10.2)
    const unsigned int lds0 = (unsigned int)(unsigned long long)(uintptr_t)&smem[0];
    tdm_load_2d_padded(lds0,                             X + (size_t)row_base * K, K, 16);
    tdm_load_2d_padded(lds0 + (unsigned int)(16 * KP * 4), W,                      K, 64);
    wait_tensorcnt0();
  }
  __syncthreads();
#else
  for (int i = threadIdx.x; i < 16 * K; i += blockDim.x)
    xs[(i / K) * KP + (i % K)] = X[(size_t)row_base * K + i];
  for (int i = threadIdx.x; i < 64 * K; i += blockDim.x)
    wls[(i / K) * KP + (i % K)] = W[i];
  __syncthreads();
#endif

  const float* __restrict__ xrow = xs  + l15 * KP;
  const float* __restrict__ wrow = wls + col * KP;

  v8f c = {};
  for (int k0 = 0; k0 < K; k0 += 4) {
    const int ka = k0 + half * 2;
    v2f a; a.x = xrow[ka]; a.y = xrow[ka + 1];
    v2f b; b.x = wrow[ka]; b.y = wrow[ka + 1];
    // D = A(16x4) * B(4x16) + C, fp32 end-to-end
    c = __builtin_amdgcn_wmma_f32_16x16x4_f32(false, a, false, b, (short)0, c,
                                              false, false);
  }

  const float bias = B[col];
#pragma unroll
  for (int r = 0; r < 8; ++r) {
    const int m = row_base + half * 8 + r;
    O[(size_t)m * DOUT + col] = c[r] + bias;
  }
}

// ---------------------------------------------------------------------------
// Per-edge attention logits + segment max. One wave32 per edge:
// each lane loads float2 of q[dst], k[src]; shuffle-reduce 64-dim dot.
// ---------------------------------------------------------------------------
__global__ void edge_logits_kernel(const float* __restrict__ q, const float* __restrict__ k,
                                   const float* __restrict__ vbuf,
                                   const int* __restrict__ src, const int* __restrict__ dst,
                                   float* __restrict__ ebuf, float* __restrict__ mbuf, int E)
{
  const int lane = threadIdx.x & 31;
  const int e = blockIdx.x * (blockDim.x >> 5) + (threadIdx.x >> 5);
  if (e >= E) return;
  const int s = src[e], d = dst[e];
  // warm v[src] rows into cache for the aggregation pass (global_prefetch_b8)
  __builtin_prefetch(vbuf + (size_t)s * DOUT, 0, 1);

  const float2 qq = ((const float2*)(q + (size_t)d * DOUT))[lane];
  const float2 kk = ((const float2*)(k + (size_t)s * DOUT))[lane];
  float p = qq.x * kk.x + qq.y * kk.y;
#pragma unroll
  for (int off = 16; off > 0; off >>= 1) p += __shfl_down(p, off, 32);

  if (lane == 0) {
    const float lg = p * 0.125f;   // 1/sqrt(DOUT), DOUT=64
    ebuf[e] = lg;
    atomicMax(mbuf + d, lg);       // global_atomic_max_num_f32
  }
}

// ---------------------------------------------------------------------------
// e = exp(logit - m[dst]); s[dst] += e. One thread per edge.
// ---------------------------------------------------------------------------
__global__ void edge_expsum_kernel(float* __restrict__ ebuf, const float* __restrict__ mbuf,
                                   float* __restrict__ sbuf, const int* __restrict__ dst, int E)
{
  const int e = blockIdx.x * blockDim.x + threadIdx.x;
  if (e >= E) return;
  const int d = dst[e];
  float mm = mbuf[d];
  if (!(mm > -INFINITY && mm < INFINITY)) mm = 0.0f;   // match reference finite-fix
  const float ev = __expf(ebuf[e] - mm);
  ebuf[e] = ev;
  atomicAdd(sbuf + d, ev);
}

// ---------------------------------------------------------------------------
// out[dst] += alpha * v[src], alpha = e / (s[dst] + 1e-16).
// One wave32 per edge; 2 fp32 atomic adds per lane. The accumulator already
// holds the skip projection, so agg+skip needs no extra pass.
// ---------------------------------------------------------------------------
__global__ void edge_agg_kernel(const float* __restrict__ ebuf, const float* __restrict__ sbuf,
                                const float* __restrict__ v, const int* __restrict__ src,
                                const int* __restrict__ dst, float* __restrict__ out, int E)
{
  const int lane = threadIdx.x & 31;
  const int e = blockIdx.x * (blockDim.x >> 5) + (threadIdx.x >> 5);
  if (e >= E) return;
  const int s = src[e], d = dst[e];
  const float alpha = ebuf[e] / (sbuf[d] + 1e-16f);
  const float2 vv = ((const float2*)(v + (size_t)s * DOUT))[lane];
  atomicAdd(out + (size_t)d * DOUT + 2 * lane,     alpha * vv.x);
  atomicAdd(out + (size_t)d * DOUT + 2 * lane + 1, alpha * vv.y);
}

// ---------------------------------------------------------------------------
// elementwise ReLU between layers
// ---------------------------------------------------------------------------
__global__ void relu_kernel(float* __restrict__ h, int n) {
  int i = blockIdx.x * blockDim.x + threadIdx.x;
  if (i < n) h[i] = fmaxf(h[i], 0.0f);
}

// ---------------------------------------------------------------------------
extern "C" void kernel_launch(void* const* d_in, const int* in_sizes, int n_in,
                              void* d_out, int out_size, void* d_ws, size_t ws_size,
                              hipStream_t stream)
{
  const float* x   = (const float*)d_in[0];
  const int*   ei  = (const int*)  d_in[1];
  const float *Wq1 = (const float*)d_in[2],  *bq1 = (const float*)d_in[3];
  const float *Wk1 = (const float*)d_in[4],  *bk1 = (const float*)d_in[5];
  const float *Wv1 = (const float*)d_in[6],  *bv1 = (const float*)d_in[7];
  const float *Ws1 = (const float*)d_in[8],  *bs1 = (const float*)d_in[9];
  const float *Wq2 = (const float*)d_in[10], *bq2 = (const float*)d_in[11];
  const float *Wk2 = (const float*)d_in[12], *bk2 = (const float*)d_in[13];
  const float *Wv2 = (const float*)d_in[14], *bv2 = (const float*)d_in[15];
  const float *Ws2 = (const float*)d_in[16], *bs2 = (const float*)d_in[17];

  const int N = in_sizes[0] / DIN;   // 50000
  const int E = in_sizes[1] / 2;     // 800000
  const int* src = ei;               // edge_index[0]
  const int* dst = ei + E;           // edge_index[1]

  // workspace layout (floats): q | k | v | h | edge | m | s  (~55 MB)
  float* ws = (float*)d_ws;
  float* q  = ws;
  float* k  = q  + (size_t)N * DOUT;
  float* v  = k  + (size_t)N * DOUT;
  float* h  = v  + (size_t)N * DOUT;   // layer-1 accumulator, then relu'd input
  float* eb = h  + (size_t)N * DOUT;   // per-edge logits / exp values
  float* mb = eb + (size_t)E;          // segment max
  float* sb = mb + (size_t)N;          // segment sum
  float* out = (float*)d_out;

  const dim3 projGrid((N + 15) / 16, 4);
  const dim3 projBlock(128);
  const int  WPB = 256 / 32;                       // waves per block in edge kernels
  const int  edgeWaveBlocks = (E + WPB - 1) / WPB;
  const int  edgeThrBlocks  = (E + 255) / 256;
  const int  nodeBlocks     = (N + 255) / 256;
  const int  featBlocks     = (N * DOUT + 255) / 256;

  // ---------------- layer 1 (DIN -> DOUT) ----------------
  init_ms_kernel<<<nodeBlocks, 256, 0, stream>>>(mb, sb, N);
  proj4_wmma_kernel<<<projGrid, projBlock, 0, stream>>>(x, DIN,
      Wq1, bq1, q,  Wk1, bk1, k,  Wv1, bv1, v,  Ws1, bs1, h);
  edge_logits_kernel<<<edgeWaveBlocks, 256, 0, stream>>>(q, k, v, src, dst, eb, mb, E);
  edge_expsum_kernel<<<edgeThrBlocks, 256, 0, stream>>>(eb, mb, sb, dst, E);
  edge_agg_kernel<<<edgeWaveBlocks, 256, 0, stream>>>(eb, sb, v, src, dst, h, E);
  relu_kernel<<<featBlocks, 256, 0, stream>>>(h, N * DOUT);

  // ---------------- layer 2 (DOUT -> DOUT) ----------------
  init_ms_kernel<<<nodeBlocks, 256, 0, stream>>>(mb, sb, N);
  proj4_wmma_kernel<<<projGrid, projBlock, 0, stream>>>(h, DOUT,
      Wq2, bq2, q,  Wk2, bk2, k,  Wv2, bv2, v,  Ws2, bs2, out);
  edge_logits_kernel<<<edgeWaveBlocks, 256, 0, stream>>>(q, k, v, src, dst, eb, mb, E);
  edge_expsum_kernel<<<edgeThrBlocks, 256, 0, stream>>>(eb, mb, sb, dst, E);
  edge_agg_kernel<<<edgeWaveBlocks, 256, 0, stream>>>(eb, sb, v, src, dst, out, E);
}